// DRAGN_75634374083017
// MI455X (gfx1250) — compile-verified
//
#include <hip/hip_runtime.h>
#include <math.h>

typedef float v2f __attribute__((ext_vector_type(2)));
typedef float v8f __attribute__((ext_vector_type(8)));

#define NBATCH 128
#define NSLICE 13
#define NS_USED 12      // last slice is never referenced by the reference windows
#define HH 64
#define WW 512
#define KK 10

#define TW 32           // output tile width (columns per block)
#define TC 40           // tile cols incl. 4-column halo each side
#define SLICE_F (HH*TC) // floats per slice buffer (2560)

// Compute a 4-row x 2-col patch of one conv output slice from LDS buffer `inb`.
// out rows ys..ys+3, cols c..c+1 (buffer coords). Window slices j..j+2.
__device__ __forceinline__ void conv_tile(const float* __restrict__ inb,
                                          int j, int ys, int c,
                                          const float* __restrict__ wk,
                                          v2f* acc)
{
    v2f z = {0.f, 0.f};
    acc[0] = z; acc[1] = z; acc[2] = z; acc[3] = z;
#pragma unroll
    for (int cc = 0; cc < 3; ++cc) {
        const float* P = inb + (j + cc) * SLICE_F;
        float r[6][4];
#pragma unroll
        for (int rr = 0; rr < 6; ++rr) {
            int yy = ys - 1 + rr;
            if (yy >= 0 && yy < HH) {
                const float* row = P + yy * TC + (c - 1);
                r[rr][0] = row[0]; r[rr][1] = row[1];
                r[rr][2] = row[2]; r[rr][3] = row[3];
            } else {
                r[rr][0] = 0.f; r[rr][1] = 0.f; r[rr][2] = 0.f; r[rr][3] = 0.f;
            }
        }
#pragma unroll
        for (int dy = 0; dy < 3; ++dy) {
#pragma unroll
            for (int dx = 0; dx < 3; ++dx) {
                float w = wk[(cc * 3 + dy) * 3 + dx];
                v2f wv = {w, w};
#pragma unroll
                for (int q = 0; q < 4; ++q) {
                    v2f v = {r[q + dy][dx], r[q + dy][dx + 1]};
                    acc[q] += wv * v;   // -> v_pk_fma_f32
                }
            }
        }
    }
}

// One aggregation stage entirely in LDS. Stage index s=1..3 shrinks the valid
// halo; columns whose global x is outside [0,W) are forced to zero to match
// the reference's per-stage SAME zero padding.
__device__ __forceinline__ void stage(const float* __restrict__ inb,
                                      float* __restrict__ outb,
                                      int n_out, int s,
                                      const float* __restrict__ wk,
                                      int x0, int tid, int nthr)
{
    const int cLo = s;
    const int vc  = TC - 2 * s;   // even
    const int npc = vc >> 1;
    const int per_slice = (HH / 4) * npc;
    const int total = n_out * per_slice;
    for (int t = tid; t < total; t += nthr) {
        int j   = t / per_slice;
        int rem = t - j * per_slice;
        int ys  = (rem / npc) << 2;
        int c   = cLo + ((rem - (rem / npc) * npc) << 1);
        v2f acc[4];
        conv_tile(inb, j, ys, c, wk, acc);
        int gx = x0 - 4 + c;
        float mx = (gx     >= 0 && gx     < WW) ? 1.f : 0.f;
        float my = (gx + 1 >= 0 && gx + 1 < WW) ? 1.f : 0.f;
        float* Q = outb + j * SLICE_F + ys * TC + c;
#pragma unroll
        for (int q = 0; q < 4; ++q) {
            Q[q * TC]     = acc[q].x * mx;
            Q[q * TC + 1] = acc[q].y * my;
        }
    }
}

extern "C" __global__ void __launch_bounds__(512)
dragn_conv_pipeline(const float* __restrict__ x,
                    const float* __restrict__ cw,
                    float* __restrict__ gfv)
{
    extern __shared__ float smem[];
    float* A = smem;                        // 12 slices
    float* B = smem + NS_USED * SLICE_F;    // 10 slices
    const int x0   = blockIdx.x * TW;
    const int b    = blockIdx.y;
    const int tid  = threadIdx.x;
    const int nthr = blockDim.x;

    float wk[27];
#pragma unroll
    for (int i = 0; i < 27; ++i) wk[i] = cw[i];

    // Global -> LDS: 12 slices, 64 rows, 40 cols (zero-filled outside image)
    const int totalLoad = NS_USED * SLICE_F;
    for (int idx = tid; idx < totalLoad; idx += nthr) {
        int s   = idx / SLICE_F;
        int rem = idx - s * SLICE_F;
        int y   = rem / TC;
        int c   = rem - y * TC;
        int gx  = x0 - 4 + c;
        float v = 0.f;
        if (gx >= 0 && gx < WW)
            v = x[(((size_t)b * NSLICE + s) * HH + y) * WW + gx];
        A[idx] = v;
    }
    __syncthreads();

    stage(A, B, 10, 1, wk, x0, tid, nthr);  // 12 -> 10 (slice 12 of x unused)
    __syncthreads();
    stage(B, A, 7, 2, wk, x0, tid, nthr);   // 10 -> 7
    __syncthreads();
    stage(A, B, 4, 3, wk, x0, tid, nthr);   // 7 -> 4
    __syncthreads();

    // stage 4: 4 -> 1, write straight to global gfv (cols are all in-image)
    const int npc4 = TW >> 1;                  // 16 column pairs
    const int total4 = (HH / 4) * npc4;        // 256 tiles
    for (int t = tid; t < total4; t += nthr) {
        int ys = (t / npc4) << 2;
        int c  = 4 + ((t - (t / npc4) * npc4) << 1);
        v2f acc[4];
        conv_tile(B, 0, ys, c, wk, acc);
        float* q = gfv + ((size_t)b * HH + ys) * WW + (x0 - 4 + c);
#pragma unroll
        for (int r = 0; r < 4; ++r) {
            q[(size_t)r * WW]     = acc[r].x;
            q[(size_t)r * WW + 1] = acc[r].y;
        }
    }
}

// Projection: out[m, n] = sigmoid( sum_k gfv[m, k] * pw[n, k] + pb[n] )
// M = 8192 (b*h), N = 10 (padded to 16), K = 512. One 16x16 tile per wave
// via V_WMMA_F32_16X16X4_F32 (native f32, exact accumulation).
// A 16x4 layout: lanes 0-15 -> {K,K+1}, lanes 16-31 -> {K+2,K+3} (row = lane&15).
// B stored as A-layout of B^T. C/D: VGPR r -> row (lane>>4)*8 + r, col lane&15.
extern "C" __global__ void __launch_bounds__(256)
dragn_proj_wmma(const float* __restrict__ gfv,
                const float* __restrict__ pw,
                const float* __restrict__ pb,
                float* __restrict__ out)
{
    const int lane = threadIdx.x & 31;
    const int tile = blockIdx.x * (blockDim.x >> 5) + (threadIdx.x >> 5);
    const int m0   = tile * 16;
    const int half = lane >> 4;
    const int lr   = lane & 15;

    const v2f* arow = (const v2f*)(gfv + (size_t)(m0 + lr) * WW);
    const int  nsel  = (lr < KK) ? lr : 0;
    const float msk  = (lr < KK) ? 1.f : 0.f;
    const v2f  bmask = {msk, msk};
    const v2f* brow  = (const v2f*)(pw + (size_t)nsel * WW);

    v8f acc0 = {0,0,0,0,0,0,0,0};
    v8f acc1 = {0,0,0,0,0,0,0,0};
    v8f acc2 = {0,0,0,0,0,0,0,0};
    v8f acc3 = {0,0,0,0,0,0,0,0};

    for (int k = 0; k < WW; k += 16) {
        int base = (k >> 1) + half;            // v2f index: K = k + 2*half
        v2f a0 = arow[base];     v2f b0 = brow[base]     * bmask;
        v2f a1 = arow[base + 2]; v2f b1 = brow[base + 2] * bmask;
        v2f a2 = arow[base + 4]; v2f b2 = brow[base + 4] * bmask;
        v2f a3 = arow[base + 6]; v2f b3 = brow[base + 6] * bmask;
        acc0 = __builtin_amdgcn_wmma_f32_16x16x4_f32(false, a0, false, b0, (short)0, acc0, false, false);
        acc1 = __builtin_amdgcn_wmma_f32_16x16x4_f32(false, a1, false, b1, (short)0, acc1, false, false);
        acc2 = __builtin_amdgcn_wmma_f32_16x16x4_f32(false, a2, false, b2, (short)0, acc2, false, false);
        acc3 = __builtin_amdgcn_wmma_f32_16x16x4_f32(false, a3, false, b3, (short)0, acc3, false, false);
    }
    v8f acc = (acc0 + acc1) + (acc2 + acc3);

    if (lr < KK) {
        float bias = pb[lr];
#pragma unroll
        for (int r = 0; r < 8; ++r) {
            int m = m0 + half * 8 + r;
            float v = acc[r] + bias;
            out[(size_t)m * KK + lr] = 1.f / (1.f + __expf(-v));
        }
    }
}

extern "C" void kernel_launch(void* const* d_in, const int* in_sizes, int n_in,
                              void* d_out, int out_size, void* d_ws, size_t ws_size,
                              hipStream_t stream)
{
    const float* x  = (const float*)d_in[0];   // (128,13,64,512) f32
    const float* cw = (const float*)d_in[1];   // (1,3,3,3) f32
    const float* pw = (const float*)d_in[2];   // (10,512) f32
    const float* pb = (const float*)d_in[3];   // (10,) f32
    float* out = (float*)d_out;                // (128,64,10) f32
    float* gfv = (float*)d_ws;                 // 128*64*512 f32 = 16.78 MB scratch

    const size_t ldsBytes = (size_t)(NS_USED + 10) * SLICE_F * sizeof(float); // 225,280 B
    dim3 cgrid(WW / TW, NBATCH);  // (16, 128)
    dragn_conv_pipeline<<<cgrid, 512, ldsBytes, stream>>>(x, cw, gfv);

    const int tiles = (NBATCH * HH) / 16;  // 512 wave-tiles
    const int wavesPerBlk = 256 / 32;      // 8
    dragn_proj_wmma<<<dim3(tiles / wavesPerBlk), 256, 0, stream>>>(gfv, pw, pb, out);
}